// Attention_644245094787
// MI455X (gfx1250) — compile-verified
//
#include <hip/hip_runtime.h>
#include <math.h>

typedef float v2f __attribute__((ext_vector_type(2)));
typedef float v8f __attribute__((ext_vector_type(8)));

#define B_ 2
#define H_ 8
#define L_ 2048
#define D_ 64
#define M_ 128
#define CHUNK 64
#define NC (L_ / CHUNK)   // 32 chunks per (b,h)
#define LT (L_ / 16)      // 128 16-row tiles per (b,h)

#define DATA_NORM 0.35355339059327373f  // 1/d^(1/4), d=64
#define RATIO_    0.08838834764831845f  // 1/sqrt(M), M=128
#define EPS_      1e-6f

__device__ __forceinline__ v8f wmma4(v2f a, v2f b, v8f c) {
  // D = A(16x4,f32) * B(4x16,f32) + C(16x16,f32)
  return __builtin_amdgcn_wmma_f32_16x16x4_f32(false, a, false, b, (short)0, c,
                                               false, false);
}

__device__ __forceinline__ v8f v8zero() {
  v8f z = {0.f, 0.f, 0.f, 0.f, 0.f, 0.f, 0.f, 0.f};
  return z;
}

// ---------------------------------------------------------------------------
// Kernel 1: projection + softmax-kernel transform.
// One block = 16 sequence rows x all M=128 cols. 8 waves, one 16x16 WMMA
// C-tile per wave, K-loop over D=64 in steps of 4.
// is_query=1: apply per-row max stabilizer + exp, write q' directly.
// is_query=0: write raw dash + diag + per-block max (key stab is global).
// ---------------------------------------------------------------------------
__global__ __launch_bounds__(256)
void favor_proj_kernel(const float* __restrict__ data,
                       const float* __restrict__ proj,
                       float* __restrict__ prime_out,
                       float* __restrict__ dash_out,
                       float* __restrict__ diag_out,
                       float* __restrict__ partmax_out,
                       int is_query)
{
  __shared__ float sDash[16][M_ + 1];
  __shared__ float sRmax[16];
  __shared__ float sDiag[16];

  const int lt = blockIdx.x % LT;
  const int bh = blockIdx.x / LT;
  const int b  = bh >> 3;
  const int h  = bh & 7;
  const int l0 = lt * 16;

  const int lane = threadIdx.x & 31;
  const int wave = threadIdx.x >> 5;
  const int r    = lane & 15;          // row/col within tile
  const int koff = (lane >> 4) << 1;   // 0 or 2 : K-halves of the 16x4 frag
  const int ci   = (lane >> 4) << 3;   // 0 or 8 : C-row offset
  const int m0   = wave * 16;          // this wave's output column tile

  const float* __restrict__ arow =
      data + (((size_t)b * L_ + (l0 + r)) * H_ + h) * D_;
  const float* __restrict__ brow = proj + (size_t)(m0 + r) * D_;

  v8f acc = v8zero();
#pragma unroll
  for (int kb = 0; kb < D_; kb += 4) {
    v2f a  = { DATA_NORM * arow[kb + koff], DATA_NORM * arow[kb + koff + 1] };
    v2f bb = { brow[kb + koff], brow[kb + koff + 1] };   // B[d][m] = P[m][d]
    acc = wmma4(a, bb, acc);
  }
#pragma unroll
  for (int vv = 0; vv < 8; ++vv)
    sDash[vv + ci][m0 + r] = acc[vv];
  __syncthreads();

  const int t = threadIdx.x;
  if (t < 16) {
    float mx = -3.4e38f;
    for (int m = 0; m < M_; ++m) mx = fmaxf(mx, sDash[t][m]);
    sRmax[t] = mx;
    const float* dr = data + (((size_t)b * L_ + (l0 + t)) * H_ + h) * D_;
    float ss = 0.f;
    for (int d = 0; d < D_; ++d) { float x = dr[d]; ss += x * x; }
    sDiag[t] = 0.5f * DATA_NORM * DATA_NORM * ss;
  }
  __syncthreads();

  if (is_query) {
#pragma unroll
    for (int kk = 0; kk < 8; ++kk) {
      int idx = t + kk * 256;
      int row = idx >> 7, col = idx & (M_ - 1);
      float val = RATIO_ * (expf(sDash[row][col] - sDiag[row] - sRmax[row]) + EPS_);
      prime_out[((size_t)bh * L_ + (l0 + row)) * M_ + col] = val;
    }
  } else {
#pragma unroll
    for (int kk = 0; kk < 8; ++kk) {
      int idx = t + kk * 256;
      int row = idx >> 7, col = idx & (M_ - 1);
      dash_out[((size_t)bh * L_ + (l0 + row)) * M_ + col] = sDash[row][col];
    }
    if (t < 16) diag_out[(size_t)bh * L_ + l0 + t] = sDiag[t];
    if (t == 0) {
      float mx = -3.4e38f;
      for (int i = 0; i < 16; ++i) mx = fmaxf(mx, sRmax[i]);
      partmax_out[bh * LT + lt] = mx;
    }
  }
}

// ---------------------------------------------------------------------------
// Kernel 2: reduce per-tile maxima -> per-(b,h) key stabilizer.
// ---------------------------------------------------------------------------
__global__ void favor_kmax_reduce(const float* __restrict__ partmax,
                                  float* __restrict__ stab)
{
  __shared__ float sm[LT];
  int bh = blockIdx.x, t = threadIdx.x;
  sm[t] = partmax[bh * LT + t];
  __syncthreads();
  for (int s = LT >> 1; s > 0; s >>= 1) {
    if (t < s) sm[t] = fmaxf(sm[t], sm[t + s]);
    __syncthreads();
  }
  if (t == 0) stab[bh] = sm[0];
}

// ---------------------------------------------------------------------------
// Kernel 3: elementwise key transform in place: k' = ratio*(exp(dash-diag-stab)+eps)
// ---------------------------------------------------------------------------
__global__ __launch_bounds__(256)
void favor_ktransform(float* __restrict__ kbuf,
                      const float* __restrict__ diagk,
                      const float* __restrict__ stab)
{
  size_t idx = (size_t)blockIdx.x * 256 + threadIdx.x;
  size_t rl  = idx >> 7;        // bh*L + l   (M=128)
  int bh     = (int)(rl >> 11); // L=2048
  kbuf[idx]  = RATIO_ * (expf(kbuf[idx] - diagk[rl] - stab[bh]) + EPS_);
}

// ---------------------------------------------------------------------------
// Kernel 4: per-chunk dKV = K'^T V  (and dksum = colsum K').
// 512 independent blocks (16 bh x 32 chunks); 8 waves x 4 WMMA D-tiles each.
// ---------------------------------------------------------------------------
__global__ __launch_bounds__(256)
void favor_chunk_kv(const float* __restrict__ kprime,
                    const float* __restrict__ value,
                    float* __restrict__ kvd,
                    float* __restrict__ ksumd)
{
  const int c  = blockIdx.x % NC;
  const int bh = blockIdx.x / NC;
  const int b = bh >> 3, h = bh & 7;
  const int t0 = c * CHUNK;

  const int lane = threadIdx.x & 31;
  const int wave = threadIdx.x >> 5;
  const int r = lane & 15;
  const int koff = (lane >> 4) << 1;
  const int ci = (lane >> 4) << 3;
  const int m0 = wave * 16;            // this wave's 16 rows of the M dim

  const float* __restrict__ kp = kprime + (size_t)bh * L_ * M_;

  __builtin_prefetch(value + (((size_t)b * L_ + t0) * H_ + h) * D_, 0, 0);

  v8f a0 = v8zero(), a1 = v8zero(), a2 = v8zero(), a3 = v8zero();
#pragma unroll 4
  for (int jb = 0; jb < CHUNK; jb += 4) {
    int j0 = t0 + jb + koff;
    // A[m][j] = K'[t0+j][m]
    v2f af = { kp[(size_t)j0 * M_ + m0 + r],
               kp[(size_t)(j0 + 1) * M_ + m0 + r] };
    const float* v0 = value + (((size_t)b * L_ + j0) * H_ + h) * D_;
    const float* v1 = value + (((size_t)b * L_ + j0 + 1) * H_ + h) * D_;
    v2f b0 = { v0[r],      v1[r]      }; a0 = wmma4(af, b0, a0);
    v2f b1 = { v0[16 + r], v1[16 + r] }; a1 = wmma4(af, b1, a1);
    v2f b2 = { v0[32 + r], v1[32 + r] }; a2 = wmma4(af, b2, a2);
    v2f b3 = { v0[48 + r], v1[48 + r] }; a3 = wmma4(af, b3, a3);
  }
  float* base = kvd + (size_t)blockIdx.x * (M_ * D_);
#pragma unroll
  for (int vv = 0; vv < 8; ++vv) {
    int m = m0 + vv + ci;
    base[(size_t)m * D_ + r]      = a0[vv];
    base[(size_t)m * D_ + 16 + r] = a1[vv];
    base[(size_t)m * D_ + 32 + r] = a2[vv];
    base[(size_t)m * D_ + 48 + r] = a3[vv];
  }
  if (threadIdx.x < M_) {
    int m = threadIdx.x;
    float s = 0.f;
    for (int j = 0; j < CHUNK; ++j) s += kp[(size_t)(t0 + j) * M_ + m];
    ksumd[(size_t)blockIdx.x * M_ + m] = s;
  }
}

// ---------------------------------------------------------------------------
// Kernel 5: exclusive prefix-sum of chunk states over NC chunks per (b,h).
// KV state (128x64) distributed as 32 registers/thread; pure streaming.
// ---------------------------------------------------------------------------
__global__ __launch_bounds__(256)
void favor_scan(float* __restrict__ kvd, float* __restrict__ ksumd)
{
  const int bh = blockIdx.x;
  const int t = threadIdx.x;
  float run[32];
#pragma unroll
  for (int i = 0; i < 32; ++i) run[i] = 0.f;
  for (int c = 0; c < NC; ++c) {
    float* base = kvd + (size_t)(bh * NC + c) * (M_ * D_);
#pragma unroll
    for (int i = 0; i < 32; ++i) {
      int idx = t + (i << 8);
      float d = base[idx];
      base[idx] = run[i];       // exclusive prefix: state BEFORE chunk c
      run[i] += d;
    }
  }
  if (t < M_) {
    float rs = 0.f;
    for (int c = 0; c < NC; ++c) {
      size_t idx = (size_t)(bh * NC + c) * M_ + t;
      float d = ksumd[idx];
      ksumd[idx] = rs;
      rs += d;
    }
  }
}

// ---------------------------------------------------------------------------
// Kernel 6: per-chunk output.
//   S = tril(Q'c K'c^T);  den = Q'c.ksum_prev + rowsum(S)
//   O = S @ Vc + Q'c @ KV_prev;  out = O / den
// ---------------------------------------------------------------------------
__global__ __launch_bounds__(256)
void favor_chunk_out(const float* __restrict__ qprime,
                     const float* __restrict__ kprime,
                     const float* __restrict__ value,
                     const float* __restrict__ kvd,
                     const float* __restrict__ ksumd,
                     float* __restrict__ out)
{
  __shared__ float sKV[M_][D_];         // 32 KB
  __shared__ float sS[CHUNK][CHUNK + 1];// 16.25 KB
  __shared__ float sKsum[M_];
  __shared__ float sDen[CHUNK];

  const int c  = blockIdx.x % NC;
  const int bh = blockIdx.x / NC;
  const int b = bh >> 3, h = bh & 7;
  const int t0 = c * CHUNK;

  const int lane = threadIdx.x & 31;
  const int wave = threadIdx.x >> 5;
  const int r = lane & 15;
  const int koff = (lane >> 4) << 1;
  const int ci = (lane >> 4) << 3;

  const float* __restrict__ qp = qprime + (size_t)bh * L_ * M_;
  const float* __restrict__ kp = kprime + (size_t)bh * L_ * M_;

  __builtin_prefetch(qp + (size_t)t0 * M_, 0, 0);
  __builtin_prefetch(value + (((size_t)b * L_ + t0) * H_ + h) * D_, 0, 0);

  {
    const float* kvbase = kvd + (size_t)blockIdx.x * (M_ * D_);
    float* dst = &sKV[0][0];
    for (int i = threadIdx.x; i < M_ * D_; i += 256) dst[i] = kvbase[i];
    if (threadIdx.x < M_)
      sKsum[threadIdx.x] = ksumd[(size_t)blockIdx.x * M_ + threadIdx.x];
  }
  __syncthreads();

  const int tm  = wave >> 1;          // 4 row tiles (chunk rows)
  const int tnb = (wave & 1) * 2;     // 2 of 4 col tiles per wave

  // ---- S = tril(Q'c @ K'c^T) -------------------------------------------
  {
    v8f s0 = v8zero(), s1 = v8zero();
    const float* qrow  = qp + (size_t)(t0 + tm * 16 + r) * M_;
    const float* krow0 = kp + (size_t)(t0 + tnb * 16 + r) * M_;
    const float* krow1 = kp + (size_t)(t0 + (tnb + 1) * 16 + r) * M_;
#pragma unroll 4
    for (int mb = 0; mb < M_; mb += 4) {
      v2f af = { qrow[mb + koff], qrow[mb + koff + 1] };
      v2f b0 = { krow0[mb + koff], krow0[mb + koff + 1] };
      s0 = wmma4(af, b0, s0);
      v2f b1 = { krow1[mb + koff], krow1[mb + koff + 1] };
      s1 = wmma4(af, b1, s1);
    }
#pragma unroll
    for (int vv = 0; vv < 8; ++vv) {
      int i  = tm * 16 + vv + ci;
      int j0 = tnb * 16 + r;
      int j1 = j0 + 16;
      sS[i][j0] = (j0 <= i) ? s0[vv] : 0.f;   // causal mask within chunk
      sS[i][j1] = (j1 <= i) ? s1[vv] : 0.f;
    }
  }
  __syncthreads();

  // ---- den --------------------------------------------------------------
  if (threadIdx.x < CHUNK) {
    int i = threadIdx.x;
    const float* qrow = qp + (size_t)(t0 + i) * M_;
    float d = 0.f;
    for (int m = 0; m < M_; ++m) d += qrow[m] * sKsum[m];
    for (int j = 0; j < CHUNK; ++j) d += sS[i][j];
    sDen[i] = d;
  }
  __syncthreads();

  // ---- O = S@Vc + Q'c@KV_prev ; write out ------------------------------
  {
    const int om = tm, onb = tnb;
    v8f o0 = v8zero(), o1 = v8zero();
    // intra-chunk: S (LDS, A-frag) x Vc (global, B-frag)
#pragma unroll 4
    for (int jb = 0; jb < CHUNK; jb += 4) {
      v2f af = { sS[om * 16 + r][jb + koff], sS[om * 16 + r][jb + koff + 1] };
      const float* v0 = value + (((size_t)b * L_ + t0 + jb + koff) * H_ + h) * D_;
      const float* v1 = value + (((size_t)b * L_ + t0 + jb + koff + 1) * H_ + h) * D_;
      v2f b0 = { v0[onb * 16 + r], v1[onb * 16 + r] };
      o0 = wmma4(af, b0, o0);
      v2f b1 = { v0[(onb + 1) * 16 + r], v1[(onb + 1) * 16 + r] };
      o1 = wmma4(af, b1, o1);
    }
    // inter-chunk: Q'c (global, A-frag) x KV_prev (LDS, B-frag)
    const float* qrow = qp + (size_t)(t0 + om * 16 + r) * M_;
#pragma unroll 4
    for (int mb = 0; mb < M_; mb += 4) {
      v2f af = { qrow[mb + koff], qrow[mb + koff + 1] };
      v2f b0 = { sKV[mb + koff][onb * 16 + r], sKV[mb + koff + 1][onb * 16 + r] };
      o0 = wmma4(af, b0, o0);
      v2f b1 = { sKV[mb + koff][(onb + 1) * 16 + r],
                 sKV[mb + koff + 1][(onb + 1) * 16 + r] };
      o1 = wmma4(af, b1, o1);
    }
#pragma unroll
    for (int vv = 0; vv < 8; ++vv) {
      int i = om * 16 + vv + ci;
      float inv = 1.0f / sDen[i];
      float* orow = out + (((size_t)b * L_ + t0 + i) * H_ + h) * D_;
      orow[onb * 16 + r]       = o0[vv] * inv;
      orow[(onb + 1) * 16 + r] = o1[vv] * inv;
    }
  }
}

// ---------------------------------------------------------------------------
extern "C" void kernel_launch(void* const* d_in, const int* in_sizes, int n_in,
                              void* d_out, int out_size, void* d_ws, size_t ws_size,
                              hipStream_t stream)
{
  (void)in_sizes; (void)n_in; (void)out_size; (void)ws_size;

  const float* q    = (const float*)d_in[0];
  const float* k    = (const float*)d_in[1];
  const float* v    = (const float*)d_in[2];
  const float* proj = (const float*)d_in[3];
  float* out = (float*)d_out;

  float* ws = (float*)d_ws;
  const size_t PRIME = (size_t)B_ * H_ * L_ * M_;          // 4,194,304 floats
  float* qprime  = ws;                                      // [B,H,L,M]
  float* kbuf    = qprime + PRIME;                          // dash -> k' in place
  float* diagk   = kbuf + PRIME;                            // [B,H,L]
  float* partmax = diagk + (size_t)B_ * H_ * L_;            // [B*H, LT]
  float* stabk   = partmax + (size_t)B_ * H_ * LT;          // [B*H]
  float* kvd     = stabk + 64;                              // [B*H*NC, M, D]
  float* ksumd   = kvd + (size_t)B_ * H_ * NC * M_ * D_;    // [B*H*NC, M]

  dim3 blk(256);
  const int projBlocks = B_ * H_ * LT;   // 2048

  favor_proj_kernel<<<projBlocks, blk, 0, stream>>>(q, proj, qprime,
                                                    nullptr, nullptr, nullptr, 1);
  favor_proj_kernel<<<projBlocks, blk, 0, stream>>>(k, proj, nullptr,
                                                    kbuf, diagk, partmax, 0);
  favor_kmax_reduce<<<B_ * H_, LT, 0, stream>>>(partmax, stabk);
  favor_ktransform<<<(int)(PRIME / 256), blk, 0, stream>>>(kbuf, diagk, stabk);
  favor_chunk_kv<<<B_ * H_ * NC, blk, 0, stream>>>(kbuf, v, kvd, ksumd);
  favor_scan<<<B_ * H_, blk, 0, stream>>>(kvd, ksumd);
  favor_chunk_out<<<B_ * H_ * NC, blk, 0, stream>>>(qprime, kbuf, v, kvd, ksumd, out);
}